// Ive_14267881357866
// MI455X (gfx1250) — compile-verified
//
#include <hip/hip_runtime.h>
#include <hip/hip_bf16.h>

typedef float v4f __attribute__((ext_vector_type(4)));

// i0e(x) = exp(-|x|) * I0(x), A&S 9.8.1/9.8.2 polynomials (cephes/scipy).
// Branchless: evaluate both regimes, select with v_cndmask.
__device__ __forceinline__ float i0e_one(float x) {
    float ax = __builtin_fabsf(x);

    // ---- small-|x| branch: exp(-ax) * P((ax/3.75)^2), ax <= 3.75 ----
    float u  = ax * 0.2666666666666667f;     // ax / 3.75
    float t2 = u * u;
    float ps = 0.0045813f;
    ps = __builtin_fmaf(ps, t2, 0.0360768f);
    ps = __builtin_fmaf(ps, t2, 0.2659732f);
    ps = __builtin_fmaf(ps, t2, 1.2067492f);
    ps = __builtin_fmaf(ps, t2, 3.0899424f);
    ps = __builtin_fmaf(ps, t2, 3.5156229f);
    ps = __builtin_fmaf(ps, t2, 1.0f);
    // exp(-ax) = 2^(-ax * log2(e))  -> v_exp_f32 (TRANS, co-executes w/ VALU)
    float e = __builtin_amdgcn_exp2f(ax * -1.4426950408889634f);
    float small_v = e * ps;

    // ---- large-|x| branch: Q(3.75/ax) / sqrt(ax), ax clamped to >= 3.75 ----
    float axs = fmaxf(ax, 3.75f);
    float s   = __builtin_amdgcn_rsqf(axs);  // v_rsq_f32: 1/sqrt(axs)
    float t   = 3.75f * (s * s);             // 3.75/axs without a v_rcp
    float pl = 0.00392377f;
    pl = __builtin_fmaf(pl, t, -0.01647633f);
    pl = __builtin_fmaf(pl, t,  0.02635537f);
    pl = __builtin_fmaf(pl, t, -0.02057706f);
    pl = __builtin_fmaf(pl, t,  0.00916281f);
    pl = __builtin_fmaf(pl, t, -0.00157565f);
    pl = __builtin_fmaf(pl, t,  0.00225319f);
    pl = __builtin_fmaf(pl, t,  0.01328592f);
    pl = __builtin_fmaf(pl, t,  0.39894228f);
    float large_v = pl * s;

    return (ax <= 3.75f) ? small_v : large_v;
}

__global__ void __launch_bounds__(256)
i0e_kernel(const float* __restrict__ x, float* __restrict__ y, long long n) {
    const long long nv4    = n >> 2;                                   // float4 count
    const long long tid    = (long long)blockIdx.x * blockDim.x + threadIdx.x;
    const long long stride = (long long)gridDim.x * blockDim.x;

    const v4f* __restrict__ xv = (const v4f*)x;
    v4f* __restrict__ yv       = (v4f*)y;

    for (long long i = tid; i < nv4; i += stride) {
        // Prefetch next strided chunk into cache (gfx1250 global_prefetch_b8).
        __builtin_prefetch((const void*)(xv + i + stride), 0, 1);

        v4f v = __builtin_nontemporal_load(xv + i);   // global_load_b128, NT
        v4f r;
        r.x = i0e_one(v.x);
        r.y = i0e_one(v.y);
        r.z = i0e_one(v.z);
        r.w = i0e_one(v.w);
        __builtin_nontemporal_store(r, yv + i);       // global_store_b128, NT
    }

    // Scalar tail for n not divisible by 4 (not hit for 16384*4096, but safe).
    for (long long i = (nv4 << 2) + tid; i < n; i += stride) {
        y[i] = i0e_one(x[i]);
    }
}

extern "C" void kernel_launch(void* const* d_in, const int* in_sizes, int n_in,
                              void* d_out, int out_size, void* d_ws, size_t ws_size,
                              hipStream_t stream) {
    const float* z = (const float*)d_in[0];
    float* out     = (float*)d_out;
    const long long n = (long long)in_sizes[0];

    const int block = 256;                      // 8 waves of 32 on CDNA5
    long long nv4 = (n + 3) >> 2;
    // ~4 float4 iterations per thread: enough ILP to hide TRANS/mem latency,
    // enough blocks (16384 for this size) to fill every WGP many times over.
    long long want = (nv4 + (long long)block * 4 - 1) / ((long long)block * 4);
    if (want < 1) want = 1;
    if (want > 1048576) want = 1048576;
    dim3 grid((unsigned)want);

    i0e_kernel<<<grid, block, 0, stream>>>(z, out, n);
}